// DPFPCell_12223476924536
// MI455X (gfx1250) — compile-verified
//
#include <hip/hip_runtime.h>
#include <math.h>

typedef __attribute__((ext_vector_type(2))) float v2f;
typedef __attribute__((ext_vector_type(4))) float v4f;
typedef __attribute__((ext_vector_type(8))) float v8f;

#define DIM   1024
#define INNER 2048
#define BATCH 32

// ------------------------------------------------------------------
// Generic fp32 WMMA GEMM: C[M,N] = A[M,K] @ B[K,N] + bias[N]
// One wave per 16x16 output tile, K stepped by 4 (V_WMMA_F32_16X16X4_F32).
// A layout (16x4 f32): v0 = K=(0|2) by lane-half, v1 = K=(1|3).
// B layout (4x16 f32): v0 = row K=(0|2), v1 = row K=(1|3), N = lane&15.
// C/D layout: vgpr r holds M = r + 8*(lane>=16), N = lane&15.
// ------------------------------------------------------------------
__global__ void gemm_f32_wmma(const float* __restrict__ A,
                              const float* __restrict__ B,
                              const float* __restrict__ bias,
                              float* __restrict__ C,
                              int M, int N, int K) {
  const int lane = threadIdx.x & 31;
  const int wave = threadIdx.x >> 5;
  const int tile = blockIdx.x * (blockDim.x >> 5) + wave;
  const int tilesN = N >> 4;
  const int tilesTotal = (M >> 4) * tilesN;
  if (tile >= tilesTotal) return;          // wave-uniform: EXEC stays all-1s
  const int tm = tile / tilesN;
  const int tn = tile % tilesN;
  const int half = lane >> 4;
  const int l = lane & 15;

  const float* arow = A + (size_t)(tm * 16 + l) * K + 2 * half;
  const float* bcol = B + tn * 16 + l + (size_t)(2 * half) * N;

  v8f acc = {};
  for (int k0 = 0; k0 < K; k0 += 4) {
    v2f a = *(const v2f*)(arow + k0);      // A[m, k0+2h], A[m, k0+2h+1]
    v2f b;
    b.x = bcol[(size_t)k0 * N];            // B[k0+2h,   n]
    b.y = bcol[(size_t)k0 * N + N];        // B[k0+2h+1, n]
    acc = __builtin_amdgcn_wmma_f32_16x16x4_f32(
        false, a, false, b, (short)0, acc, false, false);
  }
  const float bi = bias[tn * 16 + l];
#pragma unroll
  for (int r = 0; r < 8; ++r)
    C[(size_t)(tm * 16 + r + 8 * half) * N + tn * 16 + l] = acc[r] + bi;
}

// ---------------- beta = sigmoid(x @ Wb + bb), 32 tiny dots ----------------
__global__ void beta_kernel(const float* __restrict__ x,
                            const float* __restrict__ Wb,
                            const float* __restrict__ bb,
                            float* __restrict__ beta) {
  int b = threadIdx.x;  // 32 threads
  float s = bb[0];
  for (int i = 0; i < DIM; ++i) s += x[b * DIM + i] * Wb[i];
  beta[b] = 1.0f / (1.0f + expf(-s));
}

// ---------------- dpfp: f = [relu(z), relu(-z)]; out = f * roll(f,1) -------
__global__ void dpfp_kernel(const float* __restrict__ q,
                            const float* __restrict__ k,
                            float* __restrict__ qp,
                            float* __restrict__ kp) {
  int b = blockIdx.x;
  int t = threadIdx.x;  // 256
  __shared__ float fq[INNER], fk[INNER];
  for (int j = t; j < INNER; j += 256) {
    float vq = (j < DIM) ? q[b * DIM + j] : -q[b * DIM + j - DIM];
    float vk = (j < DIM) ? k[b * DIM + j] : -k[b * DIM + j - DIM];
    fq[j] = fmaxf(vq, 0.0f);
    fk[j] = fmaxf(vk, 0.0f);
  }
  __syncthreads();
  for (int j = t; j < INNER; j += 256) {
    int jm = (j + INNER - 1) & (INNER - 1);
    qp[b * INNER + j] = fq[j] * fq[jm];
    kp[b * INNER + j] = fk[j] * fk[jm];
  }
}

__global__ void zero_kernel(float* p, int n) {
  int i = blockIdx.x * blockDim.x + threadIdx.x;
  if (i < n) p[i] = 0.0f;
}

// ---------------- Pass 1: colsum[b,j] = sum_i W, rowsum[b,i] = sum_j W -----
// 256 blocks = (32 b) x (2 j-halves of 1024 cols) x (4 i-chunks of 512 rows)
__global__ void colrow_sums(const float* __restrict__ W,
                            float* __restrict__ colsum,
                            float* __restrict__ rowsum) {
  const int bid = blockIdx.x;
  const int b  = bid >> 3;
  const int jh = (bid >> 2) & 1;
  const int ic = bid & 3;
  const int t  = threadIdx.x;  // 256
  const float* Wb = W + (size_t)b * INNER * INNER;
  const int jbase = jh * 1024 + 4 * t;

  __shared__ float rowpart[512];
  for (int i = t; i < 512; i += 256) rowpart[i] = 0.0f;
  __syncthreads();

  float c0 = 0.f, c1 = 0.f, c2 = 0.f, c3 = 0.f;
  const int i0 = ic * 512;
  for (int i = i0; i < i0 + 512; ++i) {
    v4f w = *(const v4f*)(Wb + (size_t)i * INNER + jbase);
    c0 += w.x; c1 += w.y; c2 += w.z; c3 += w.w;
    float rs = (w.x + w.y) + (w.z + w.w);
    for (int off = 16; off > 0; off >>= 1) rs += __shfl_down(rs, off, 32);
    if ((t & 31) == 0) atomicAdd(&rowpart[i - i0], rs);
  }
  __syncthreads();
  for (int i = t; i < 512; i += 256)
    atomicAdd(&rowsum[b * INNER + i0 + i], rowpart[i]);
  atomicAdd(&colsum[b * INNER + jbase + 0], c0);
  atomicAdd(&colsum[b * INNER + jbase + 1], c1);
  atomicAdd(&colsum[b * INNER + jbase + 2], c2);
  atomicAdd(&colsum[b * INNER + jbase + 3], c3);
}

// ---- dv = beta*(v - colsum*kp); S = sum(kp); outv = (rowsum + dv*S)*qp ----
__global__ void small_kernel(const float* __restrict__ v,
                             const float* __restrict__ beta,
                             const float* __restrict__ colsum,
                             const float* __restrict__ rowsum,
                             const float* __restrict__ kp,
                             const float* __restrict__ qp,
                             float* __restrict__ dv,
                             float* __restrict__ outv) {
  int b = blockIdx.x;
  int t = threadIdx.x;  // 256
  __shared__ float red[256];
  float s = 0.0f;
  for (int j = t; j < INNER; j += 256) s += kp[b * INNER + j];
  red[t] = s;
  __syncthreads();
  for (int o = 128; o > 0; o >>= 1) {
    if (t < o) red[t] += red[t + o];
    __syncthreads();
  }
  const float S = red[0];
  const float be = beta[b];
  for (int j = t; j < INNER; j += 256) {
    float d = be * (v[b * INNER + j] - colsum[b * INNER + j] * kp[b * INNER + j]);
    dv[b * INNER + j] = d;
    outv[b * INNER + j] = (rowsum[b * INNER + j] + d * S) * qp[b * INNER + j];
  }
}

// ---------------- Pass 2: W_new = W + dv (x) kp, NT-streamed ----------------
__global__ void update_kernel(const float* __restrict__ W,
                              const float* __restrict__ dv,
                              const float* __restrict__ kp,
                              float* __restrict__ Wnew) {
  size_t idx = (size_t)blockIdx.x * blockDim.x + threadIdx.x;  // one float4
  size_t j4 = idx & 511;           // INNER/4 = 512
  size_t i  = (idx >> 9) & 2047;
  size_t b  = idx >> 20;           // / (512*2048)
  float d = dv[b * INNER + i];
  v4f kpv = *(const v4f*)(kp + b * INNER + j4 * 4);
  v4f w = __builtin_nontemporal_load((const v4f*)W + idx);
  v4f r = w + d * kpv;
  __builtin_nontemporal_store(r, (v4f*)Wnew + idx);
}

extern "C" void kernel_launch(void* const* d_in, const int* in_sizes, int n_in,
                              void* d_out, int out_size, void* d_ws, size_t ws_size,
                              hipStream_t stream) {
  const float* x  = (const float*)d_in[0];
  const float* W  = (const float*)d_in[1];
  const float* Wq = (const float*)d_in[2];
  const float* bq = (const float*)d_in[3];
  const float* Wk = (const float*)d_in[4];
  const float* bk = (const float*)d_in[5];
  const float* Wv = (const float*)d_in[6];
  const float* bv = (const float*)d_in[7];
  const float* Wo = (const float*)d_in[8];
  const float* bo = (const float*)d_in[9];
  const float* Wb = (const float*)d_in[10];
  const float* bb = (const float*)d_in[11];

  float* out  = (float*)d_out;                         // [32*1024]
  float* Wnew = (float*)d_out + (size_t)BATCH * DIM;   // [32*2048*2048]

  float* ws     = (float*)d_ws;
  float* q      = ws;               // 32768
  float* k      = ws + 32768;       // 32768
  float* v      = ws + 65536;       // 65536
  float* qp     = ws + 131072;      // 65536
  float* kp     = ws + 196608;      // 65536
  float* beta   = ws + 262144;      // 32
  float* colsum = ws + 262176;      // 65536
  float* rowsum = ws + 327712;      // 65536
  float* dv     = ws + 393248;      // 65536
  float* outv   = ws + 458784;      // 65536

  // q, k, v via fp32 WMMA GEMMs (M=32)
  {
    int tiles = (BATCH / 16) * (DIM / 16);   // 128
    gemm_f32_wmma<<<dim3((tiles + 3) / 4), dim3(128), 0, stream>>>(x, Wq, bq, q, BATCH, DIM, DIM);
    gemm_f32_wmma<<<dim3((tiles + 3) / 4), dim3(128), 0, stream>>>(x, Wk, bk, k, BATCH, DIM, DIM);
  }
  {
    int tiles = (BATCH / 16) * (INNER / 16); // 256
    gemm_f32_wmma<<<dim3((tiles + 3) / 4), dim3(128), 0, stream>>>(x, Wv, bv, v, BATCH, INNER, DIM);
  }
  beta_kernel<<<1, 32, 0, stream>>>(x, Wb, bb, beta);
  dpfp_kernel<<<BATCH, 256, 0, stream>>>(q, k, qp, kp);

  // Pass 1 over W: col + row sums in one read
  zero_kernel<<<(131072 + 255) / 256, 256, 0, stream>>>(colsum, 131072);  // colsum+rowsum contiguous
  colrow_sums<<<256, 256, 0, stream>>>(W, colsum, rowsum);

  small_kernel<<<BATCH, 256, 0, stream>>>(v, beta, colsum, rowsum, kp, qp, dv, outv);

  // Pass 2: stream W -> W_new with rank-1 update
  {
    size_t total4 = (size_t)BATCH * INNER * INNER / 4;  // 33,554,432
    update_kernel<<<dim3((unsigned)(total4 / 256)), dim3(256), 0, stream>>>(W, dv, kp, Wnew);
  }

  // out = outv @ Wo + bo
  {
    int tiles = (BATCH / 16) * (DIM / 16);
    gemm_f32_wmma<<<dim3((tiles + 3) / 4), dim3(128), 0, stream>>>(outv, Wo, bo, out, BATCH, DIM, INNER);
  }
}